// LADMM_87385404604829
// MI455X (gfx1250) — compile-verified
//
#include <hip/hip_runtime.h>

typedef __attribute__((ext_vector_type(2))) float v2f;
typedef __attribute__((ext_vector_type(8))) float v8f;

#define NPAD 304          // 300 padded to 19*16
#define NREAL 300
#define BROWS 65536
#define NB (BROWS * NPAD) // elements per big [B,304] buffer
#define NTILES 19         // 304/16
#define KSTEPS 76         // 304/4
#define NPAIRS (KSTEPS * NTILES * 32)   // packed B fragment pairs = 304*304/2

__device__ __forceinline__ float softt(float v, float t) {
    float a = fmaxf(fabsf(v) - t, 0.0f);
    return copysignf(a, v);
}

// ---------------- M = A^T A + al*I + g*(2I - S - S^T); pad diag = al+2g -------
__global__ void buildM(const float* __restrict__ A, const float* __restrict__ alp,
                       const float* __restrict__ gam, float* __restrict__ Mm) {
    int id = blockIdx.x * blockDim.x + threadIdx.x;
    if (id >= NPAD * NPAD) return;
    int j = id % NPAD, i = id / NPAD;
    float al = alp[0], g = gam[0];
    float v;
    if (i < NREAL && j < NREAL) {
        float s = 0.f;
#pragma unroll
        for (int r = 0; r < 9; ++r) s += A[r * NREAL + i] * A[r * NREAL + j];
        v = s;
        if (i == j) v += al + 2.f * g;
        if (i == (j + 1) % NREAL) v -= g;
        if (i == (j + NREAL - 1) % NREAL) v -= g;
    } else {
        v = (i == j) ? (al + 2.f * g) : 0.f;  // keep lambda_min(Mpad) >= al
    }
    Mm[id] = v;
}

// ---------------- Gershgorin scale + X0 = (1/R) * I (one block) ----------------
__global__ __launch_bounds__(320) void prep_inv(const float* __restrict__ Mm,
                                                float* __restrict__ X) {
    __shared__ float sred[320];
    __shared__ float sc;
    int t = threadIdx.x;
    float rs = 0.f;
    if (t < NPAD)
        for (int j = 0; j < NPAD; ++j) rs += fabsf(Mm[t * NPAD + j]);
    sred[t] = rs;
    __syncthreads();
    if (t == 0) {
        float R = 1.f;
        for (int i = 0; i < NPAD; ++i) R = fmaxf(R, sred[i]);
        sc = 1.0f / R;
    }
    __syncthreads();
    float c = sc;
    for (int idx = t; idx < NPAD * NPAD; idx += 320) {
        int i = idx / NPAD, j = idx % NPAD;
        X[idx] = (i == j) ? c : 0.f;
    }
}

// ---------------- 304x304x304 WMMA GEMM: D = A@B (mode 0) or D = 2*C - A@B ----
__global__ __launch_bounds__(256) void gemm_sq(const float* __restrict__ Am,
                                               const float* __restrict__ Bm,
                                               const float* __restrict__ Cin,
                                               float* __restrict__ Dout, int mode) {
    int wid = blockIdx.x * 8 + (threadIdx.x >> 5);
    if (wid >= NTILES * NTILES) return;      // wave-uniform early out
    int tm = wid / NTILES, tn = wid % NTILES;
    int lane = threadIdx.x & 31, mrow = lane & 15, kh = lane >> 4;

    const float* arow = Am + (size_t)(tm * 16 + mrow) * NPAD + 2 * kh;
    const float* bbase = Bm + tn * 16 + mrow;
    v8f acc = {0.f, 0.f, 0.f, 0.f, 0.f, 0.f, 0.f, 0.f};
    for (int k0 = 0; k0 < NPAD; k0 += 4) {
        v2f a = *(const v2f*)(arow + k0);
        const float* bp = bbase + (size_t)(k0 + 2 * kh) * NPAD;
        v2f b;
        b.x = bp[0];
        b.y = bp[NPAD];
        acc = __builtin_amdgcn_wmma_f32_16x16x4_f32(false, a, false, b, (short)0,
                                                    acc, false, false);
    }
    float* orow = Dout + (size_t)(tm * 16 + 8 * kh) * NPAD + tn * 16 + mrow;
    if (mode == 0) {
#pragma unroll
        for (int r = 0; r < 8; ++r) orow[(size_t)r * NPAD] = acc[r];
    } else {
        const float* crow = Cin + (size_t)(tm * 16 + 8 * kh) * NPAD + tn * 16 + mrow;
#pragma unroll
        for (int r = 0; r < 8; ++r)
            orow[(size_t)r * NPAD] = 2.0f * crow[(size_t)r * NPAD] - acc[r];
    }
}

// ---------------- pack Minv into WMMA B-fragment order (one-time) -------------
// Bp[((kstep*19 + t)*32 + lane)] = { Minv[k][n], Minv[k+1][n] }
// with k = kstep*4 + 2*(lane>>4), n = t*16 + (lane&15)
__global__ void packB(const float* __restrict__ Minv, float* __restrict__ Bp) {
    int p = blockIdx.x * blockDim.x + threadIdx.x;
    if (p >= NPAIRS) return;
    int lane = p & 31;
    int t = (p >> 5) % NTILES;
    int kstep = p / (32 * NTILES);
    int kh = lane >> 4, mrow = lane & 15;
    int k = kstep * 4 + 2 * kh;
    int n = t * 16 + mrow;
    Bp[2 * p]     = Minv[(size_t)k * NPAD + n];
    Bp[2 * p + 1] = Minv[(size_t)(k + 1) * NPAD + n];
}

// ---------------- bA = b @ A, padded columns zeroed --------------------------
__global__ void bA_kernel(const float* __restrict__ b, const float* __restrict__ A,
                          float* __restrict__ bA) {
    int id = blockIdx.x * blockDim.x + threadIdx.x;
    if (id >= NB) return;
    int col = id % NPAD, row = id / NPAD;
    if (col >= NREAL) { bA[id] = 0.f; return; }
    const float* br = b + (size_t)row * 9;
    float s = 0.f;
#pragma unroll
    for (int r = 0; r < 9; ++r) s += br[r] * A[r * NREAL + col];
    bA[id] = s;
}

// ---------------- fused elementwise: dual update + prox + resid ---------------
// State: s1 = al*w - tau, s2 = g*u - eta.  resid = bA + s1 + s2.
__global__ void ewise(const float* __restrict__ x, int ldx,
                      const float* __restrict__ bA, float* __restrict__ s1,
                      float* __restrict__ s2, float* __restrict__ resid,
                      const float* __restrict__ alp, const float* __restrict__ gam,
                      const float* __restrict__ lamp, int first) {
    int id = blockIdx.x * blockDim.x + threadIdx.x;
    if (id >= NB) return;
    int col = id % NPAD, row = id / NPAD;
    if (col >= NREAL) { resid[id] = 0.f; s1[id] = 0.f; s2[id] = 0.f; return; }
    float al = alp[0], g = gam[0], thr = lamp[0] / g;
    const float* xr = x + (size_t)row * ldx;
    float xj = xr[col];
    float xm = xr[(col + NREAL - 1) % NREAL];   // roll(x,1,axis=1)
    float d = xm - xj;
    float u, w, ns1, ns2;
    if (first) {                                // eta = tau = 0
        u = softt(d, thr);
        w = fmaxf(xj, 0.f);
        ns1 = al * w;
        ns2 = g * u;
    } else {
        float o1 = s1[id], o2 = s2[id];
        u = softt(2.f * d - o2 / g, thr);       // d + eta_new/g, eta_new = g*d - o2
        w = fmaxf(2.f * xj - o1 / al, 0.f);     // x + tau_new/al, tau_new = al*x - o1
        ns1 = al * w - al * xj + o1;
        ns2 = g * u - g * d + o2;
    }
    s1[id] = ns1;
    s2[id] = ns2;
    resid[id] = bA[id] + ns1 + ns2;
}

// ---------------- big GEMM: Out[B x n] = resid[B x 304] @ Minv (packed) -------
// wave = 16 rows x all 19 N-tiles in registers; resid read exactly once;
// B fragments pre-packed -> one coalesced v2f load per tile per k-step.
__global__ __launch_bounds__(256) void gemm_big(const float* __restrict__ Rsd,
                                                const float* __restrict__ Bp,
                                                float* __restrict__ Out, int ldo,
                                                int nmax) {
    const int lane = threadIdx.x & 31;
    const int wave = threadIdx.x >> 5;
    const int m0 = blockIdx.x * 128 + wave * 16;
    const int mrow = lane & 15;
    const int kh = lane >> 4;

    const float* arow = Rsd + (size_t)(m0 + mrow) * NPAD + 2 * kh;
    const v2f* Bpv = (const v2f*)Bp;
    v8f acc[NTILES];
    v8f z = {0.f, 0.f, 0.f, 0.f, 0.f, 0.f, 0.f, 0.f};
#pragma unroll
    for (int t = 0; t < NTILES; ++t) acc[t] = z;

    for (int ks = 0; ks < KSTEPS; ++ks) {
        v2f a = *(const v2f*)(arow + ks * 4);             // A frag, reused 19x
        const v2f* bfrag = Bpv + (size_t)(ks * NTILES) * 32 + lane;
#pragma unroll
        for (int t = 0; t < NTILES; ++t) {
            v2f b = bfrag[t * 32];                        // coalesced b64 load
            acc[t] = __builtin_amdgcn_wmma_f32_16x16x4_f32(false, a, false, b,
                                                           (short)0, acc[t],
                                                           false, false);
        }
    }
#pragma unroll
    for (int t = 0; t < NTILES; ++t) {
        int n = t * 16 + mrow;
        if (n < nmax) {
            float* orow = Out + (size_t)(m0 + 8 * kh) * ldo + n;
#pragma unroll
            for (int r = 0; r < 8; ++r) orow[(size_t)r * ldo] = acc[t][r];
        }
    }
}

extern "C" void kernel_launch(void* const* d_in, const int* in_sizes, int n_in,
                              void* d_out, int out_size, void* d_ws, size_t ws_size,
                              hipStream_t stream) {
    const float* b   = (const float*)d_in[0];
    // d_in[1] = target (unused, shape only)
    const float* x0  = (const float*)d_in[2];
    const float* A   = (const float*)d_in[3];
    const float* alp = (const float*)d_in[4];
    const float* gam = (const float*)d_in[5];
    const float* lam = (const float*)d_in[6];
    float* out = (float*)d_out;

    // workspace layout (floats): 5 small 304x304 + 5 big [B,304]
    float* Mm  = (float*)d_ws;
    float* Xa  = Mm + NPAD * NPAD;
    float* Xb  = Xa + NPAD * NPAD;
    float* Tt  = Xb + NPAD * NPAD;
    float* Bp  = Tt + NPAD * NPAD;
    float* bA  = Bp + NPAD * NPAD;
    float* rsd = bA + NB;
    float* xbf = rsd + NB;
    float* s1  = xbf + NB;
    float* s2  = s1 + NB;

    const int TB = 256;
    const int gridSmall = (NPAD * NPAD + TB - 1) / TB;   // 361 blocks
    const int gridPack  = (NPAIRS + TB - 1) / TB;        // 181 blocks
    const int gridBig   = (NB + TB - 1) / TB;            // 77824 blocks
    const int gridSq    = (NTILES * NTILES + 7) / 8;     // 46 blocks (8 waves each)
    const int gridGemm  = BROWS / 128;                   // 512 blocks

    // 1) build M (padded, SPD, lambda_min >= alpha)
    buildM<<<gridSmall, TB, 0, stream>>>(A, alp, gam, Mm);
    // 2) X0 = I / GershgorinBound(M)
    prep_inv<<<1, 320, 0, stream>>>(Mm, Xa);
    // 3) Newton-Schulz: X <- X (2I - M X); rho0 ~ 1 - alpha/R -> ~12 doublings
    float* cur = Xa;
    float* nxt = Xb;
    for (int it = 0; it < 16; ++it) {
        gemm_sq<<<gridSq, TB, 0, stream>>>(cur, Mm, nullptr, Tt, 0);   // T = X@M
        gemm_sq<<<gridSq, TB, 0, stream>>>(Tt, cur, cur, nxt, 1);      // X' = 2X - T@X
        float* tmp = cur; cur = nxt; nxt = tmp;
    }
    // 4) pack Minv into WMMA B-fragment order for the big GEMMs
    packB<<<gridPack, TB, 0, stream>>>(cur, Bp);
    // 5) bA = b @ A (constant over iterations)
    bA_kernel<<<gridBig, TB, 0, stream>>>(b, A, bA);

    // 6) five unrolled ADMM iterations
    for (int t = 0; t < 5; ++t) {
        const float* xin = (t == 0) ? x0 : xbf;
        int ldx = (t == 0) ? NREAL : NPAD;
        ewise<<<gridBig, TB, 0, stream>>>(xin, ldx, bA, s1, s2, rsd,
                                          alp, gam, lam, t == 0);
        if (t == 4)
            gemm_big<<<gridGemm, TB, 0, stream>>>(rsd, Bp, out, NREAL, NREAL);
        else
            gemm_big<<<gridGemm, TB, 0, stream>>>(rsd, Bp, xbf, NPAD, NPAD);
    }
}